// ResidualGNNBlock_72018011619519
// MI455X (gfx1250) — compile-verified
//
#include <hip/hip_runtime.h>
#include <math.h>

typedef __attribute__((ext_vector_type(2))) float v2f;
typedef __attribute__((ext_vector_type(8))) float v8f;

#define DIM 128
#define LN_EPS 1e-5f

// ---------------------------------------------------------------------------
// Zero the scatter workspace (summed[N*128] and deg[N]).
// ---------------------------------------------------------------------------
__global__ __launch_bounds__(256) void zero_ws_kernel(float* __restrict__ p, long n) {
    long i = (long)blockIdx.x * blockDim.x + threadIdx.x;
    long stride = (long)gridDim.x * blockDim.x;
    for (; i < n; i += stride) p[i] = 0.0f;
}

// ---------------------------------------------------------------------------
// Edge scatter: one wave (32 lanes) per edge; each lane moves a float4 of the
// 128-dim feature via global_atomic_add_f32. x and summed are L2-resident
// (20.5 MB each vs 192 MB L2), so this is L2-atomic bound.
// ---------------------------------------------------------------------------
__global__ __launch_bounds__(256) void scatter_mean_accum(
    const float* __restrict__ x, const int* __restrict__ ei,
    float* __restrict__ summed, float* __restrict__ deg, int nE)
{
    int tid  = blockIdx.x * blockDim.x + threadIdx.x;
    int e    = tid >> 5;
    int lane = tid & 31;
    if (e >= nE) return;
    int src = ei[e];
    int dst = ei[nE + e];
    const float4* xs = (const float4*)(x + (long)src * DIM);
    float4 v = xs[lane];
    float* so = summed + (long)dst * DIM + lane * 4;
    atomicAdd(so + 0, v.x);
    atomicAdd(so + 1, v.y);
    atomicAdd(so + 2, v.z);
    atomicAdd(so + 3, v.w);
    if (lane == 0) atomicAdd(deg + dst, 1.0f);
}

__device__ __forceinline__ float gelu_exact(float v) {
    return 0.5f * v * (1.0f + erff(v * 0.70710678118654752440f));
}

// ---------------------------------------------------------------------------
// Fused: out = LN( x + gelu( mean_nb @ Wl^T + bl + x @ Wr^T ) )
// One wave handles 16 rows x 128 cols via V_WMMA_F32_16X16X4_F32 (exact f32).
// A-frag (16x4 f32): lane<16 -> {K=k, K=k+1} of row M=lane; lane>=16 -> {K=k+2,K=k+3}.
// B-frag (4x16): lane supplies column n = t*16 + (lane&15), same K striping.
// C/D tile: VGPR v holds row M = v + 8*(lane>=16), col N = lane&15.
// ---------------------------------------------------------------------------
__global__ __launch_bounds__(128) void sage_wmma_fused(
    const float* __restrict__ x,      const float* __restrict__ summed,
    const float* __restrict__ deg,    const float* __restrict__ Wl,
    const float* __restrict__ bl,     const float* __restrict__ Wr,
    const float* __restrict__ gamma,  const float* __restrict__ beta,
    float* __restrict__ out, int n)
{
    const int wave = threadIdx.x >> 5;
    const int lane = threadIdx.x & 31;
    const int half = lane >> 4;
    const int lr   = lane & 15;
    const long row0 = ((long)blockIdx.x * 4 + wave) * 16;

    // Row this lane feeds into the A fragments (clamped so EXEC stays full).
    long rA = row0 + lr; if (rA > (long)n - 1) rA = (long)n - 1;
    const float invdeg = 1.0f / fmaxf(deg[rA], 1.0f);
    const float* __restrict__ sRow = summed + rA * DIM;
    const float* __restrict__ xRow = x + rA * DIM;

    v8f acc[8] = {};  // 8 column tiles of 16x16 f32

    #pragma unroll 4
    for (int kk = 0; kk < DIM / 4; ++kk) {
        const int k = kk * 4 + half * 2;
        v2f aM, aX;
        aM.x = sRow[k]     * invdeg;
        aM.y = sRow[k + 1] * invdeg;
        aX.x = xRow[k];
        aX.y = xRow[k + 1];
        #pragma unroll
        for (int t = 0; t < 8; ++t) {
            const float* __restrict__ wl = Wl + (long)(t * 16 + lr) * DIM + k;
            const float* __restrict__ wr = Wr + (long)(t * 16 + lr) * DIM + k;
            v2f bL; bL.x = wl[0]; bL.y = wl[1];
            v2f bR; bR.x = wr[0]; bR.y = wr[1];
            acc[t] = __builtin_amdgcn_wmma_f32_16x16x4_f32(
                false, aM, false, bL, (short)0, acc[t], false, false);
            acc[t] = __builtin_amdgcn_wmma_f32_16x16x4_f32(
                false, aX, false, bR, (short)0, acc[t], false, false);
        }
    }

    // Epilogue: bias + exact GELU + residual, all in registers.
    float s1[8] = {}, s2[8] = {};
    #pragma unroll
    for (int t = 0; t < 8; ++t) {
        const int col = t * 16 + lr;
        const float b = bl[col];
        #pragma unroll
        for (int v = 0; v < 8; ++v) {
            long r = row0 + v + 8 * half; if (r > (long)n - 1) r = (long)n - 1;
            float val = gelu_exact(acc[t][v] + b);
            float hv  = x[r * DIM + col] + val;
            acc[t][v] = hv;
            s1[v] += hv;
            s2[v] += hv * hv;
        }
    }

    // Row-wise mean/var: reduce across the 16 lanes of each half
    // (masks 1,2,4,8 never cross the half boundary on wave32).
    float mu[8], rstd[8];
    #pragma unroll
    for (int v = 0; v < 8; ++v) {
        float a = s1[v], b = s2[v];
        #pragma unroll
        for (int m = 8; m >= 1; m >>= 1) {
            a += __shfl_xor(a, m, 32);
            b += __shfl_xor(b, m, 32);
        }
        mu[v]   = a * (1.0f / DIM);
        float var = b * (1.0f / DIM) - mu[v] * mu[v];
        rstd[v] = rsqrtf(var + LN_EPS);
    }

    // Normalize + affine + store (predicated per row).
    #pragma unroll
    for (int t = 0; t < 8; ++t) {
        const int col = t * 16 + lr;
        const float g = gamma[col], be = beta[col];
        #pragma unroll
        for (int v = 0; v < 8; ++v) {
            long r = row0 + v + 8 * half;
            if (r < n) out[r * DIM + col] = (acc[t][v] - mu[v]) * rstd[v] * g + be;
        }
    }
}

// ---------------------------------------------------------------------------
extern "C" void kernel_launch(void* const* d_in, const int* in_sizes, int n_in,
                              void* d_out, int out_size, void* d_ws, size_t ws_size,
                              hipStream_t stream)
{
    (void)n_in; (void)out_size; (void)ws_size;
    const float* x     = (const float*)d_in[0];
    const int*   ei    = (const int*)d_in[1];
    const float* Wl    = (const float*)d_in[2];
    const float* bl    = (const float*)d_in[3];
    const float* Wr    = (const float*)d_in[4];
    const float* gamma = (const float*)d_in[5];
    const float* beta  = (const float*)d_in[6];
    float* out = (float*)d_out;

    const int n = in_sizes[0] / DIM;   // 40000
    const int e = in_sizes[1] / 2;     // 640000

    float* summed = (float*)d_ws;                 // [n * 128]
    float* deg    = summed + (long)n * DIM;       // [n]

    long zn = (long)n * DIM + n;
    zero_ws_kernel<<<1024, 256, 0, stream>>>(summed, zn);

    long scatter_threads = (long)e * 32;
    int scatter_blocks = (int)((scatter_threads + 255) / 256);
    scatter_mean_accum<<<scatter_blocks, 256, 0, stream>>>(x, ei, summed, deg, e);

    const int rows_per_block = 64;  // 4 waves * 16 rows
    int blocks = (n + rows_per_block - 1) / rows_per_block;
    sage_wmma_fused<<<blocks, 128, 0, stream>>>(x, summed, deg, Wl, bl, Wr,
                                                gamma, beta, out, n);
}